// Mol2MSLoss_64355789963326
// MI455X (gfx1250) — compile-verified
//
#include <hip/hip_runtime.h>
#include <math.h>

// ---- problem constants (fixed by reference: B=32, S=2048) ----
#define SROW 2048
#define BROWS 32
#define NTOT (SROW * BROWS)

typedef float v2f  __attribute__((ext_vector_type(2)));
typedef float v8f  __attribute__((ext_vector_type(8)));
typedef unsigned int u32x4 __attribute__((ext_vector_type(4)));
typedef int   i32x8 __attribute__((ext_vector_type(8)));
typedef int   i32x4 __attribute__((ext_vector_type(4)));

#if defined(__HIP_DEVICE_COMPILE__) && __has_builtin(__builtin_amdgcn_wmma_f32_16x16x4_f32)
#define USE_WMMA 1
#else
#define USE_WMMA 0
#endif
#if defined(__HIP_DEVICE_COMPILE__) && __has_builtin(__builtin_amdgcn_tensor_load_to_lds) && \
    __has_builtin(__builtin_amdgcn_s_wait_tensorcnt)
#define USE_TDM 1
#else
#define USE_TDM 0
#endif

__device__ __forceinline__ float sgnf(float x) {
    return (x > 0.0f) ? 1.0f : ((x < 0.0f) ? -1.0f : 0.0f);
}

// ---------------------------------------------------------------------------
// Kernel 1: soft-Jaccard pairwise min term.
// grid = (16, 32), block = 256 (8 waves). Block (bx, row) handles pred cols
// [bx*128, bx*128+128) of batch row `row` against the full mz row.
// ws[row*16 + bx] = partial soft_intersect (deterministic per-block sum).
// ---------------------------------------------------------------------------
__global__ void __launch_bounds__(256)
sj_pairwise_kernel(const float* __restrict__ pred_mz,
                   const float* __restrict__ mz,
                   const float* __restrict__ pred_inten,
                   float* __restrict__ ws) {
    __shared__ __align__(16) float lds_mz[SROW];
    __shared__ int s_plen, s_mlen;
    __shared__ float s_wsum[8];

    const int tid  = threadIdx.x;
    const int lane = tid & 31;
    const int wid  = tid >> 5;
    const int row  = blockIdx.y;
    const int n    = lane & 15;                 // pred column within 16-wide group
    const int g    = blockIdx.x * 8 + wid;      // pred group id 0..127

    const float* mzrow = mz      + row * SROW;
    const float* pmrow = pred_mz + row * SROW;

    if (tid == 0) {
        s_plen = 0;
        s_mlen = 0;
#if USE_TDM
        // --- Tensor Data Mover: async copy mz row (8KB) global -> LDS ---
        unsigned long long ga = (unsigned long long)(const void*)mzrow;
        unsigned lds_off = (unsigned)(unsigned long long)(void*)&lds_mz[0];
        u32x4 g0 = { 1u,                                   // count=1, user mode
                     lds_off,                              // lds_addr
                     (unsigned)ga,                         // global_addr[31:0]
                     ((unsigned)(ga >> 32) & 0x01FFFFFFu)  // global_addr[56:32]
                     | 0x80000000u };                      // type=2 ("image")
        i32x8 g1 = { (int)(2u << 16),        // data_size=2 (4 bytes)
                     (int)(2048u << 16),     // tensor_dim0 = 2048 (bits 79:48)
                     (int)(1u << 16),        // tensor_dim1 = 1
                     (int)(2048u << 16),     // tile_dim0 = 2048
                     1,                      // tile_dim1 = 1
                     2048,                   // tensor_dim0_stride
                     0, 0 };
        i32x4 z4 = { 0, 0, 0, 0 };
        i32x8 z8 = { 0, 0, 0, 0, 0, 0, 0, 0 };
        // 6-arg form (amdgpu-toolchain / clang-23 + therock-10.0 headers)
        __builtin_amdgcn_tensor_load_to_lds(g0, g1, z4, z4, z8, 0);
#endif
    }
    __syncthreads();

    // Scan row for last-negative indices (L2-resident loads, overlaps TDM).
    for (int j = tid; j < SROW; j += 256) {
        float pv = pmrow[j];
        float mv = mzrow[j];
#if !USE_TDM
        lds_mz[j] = mv;                 // cooperative fallback copy
#endif
        if (pv < 0.0f) atomicMax(&s_plen, j);
        if (mv < 0.0f) atomicMax(&s_mlen, j);
    }
#if USE_TDM
    if (tid == 0) __builtin_amdgcn_s_wait_tensorcnt(0);
#endif
    __syncthreads();

    const int plen = s_plen;
    const int mlen = s_mlen;

    // Invalidate mz[j] for j >= mlen with a large FINITE sentinel
    // (finite so sentinel * zero-B-row == 0, never NaN inside WMMA).
    for (int j = mlen + tid; j < SROW; j += 256) lds_mz[j] = 1.0e30f;
    __syncthreads();

    // ---- main pairwise loop: D[i][j] = mz_i - pred_j via WMMA outer product.
    const float pcol = pmrow[g * 16 + n];

    v8f dmin;
#pragma unroll
    for (int r = 0; r < 8; ++r) dmin[r] = 3.0e38f;

#if USE_WMMA
    v8f c;
#pragma unroll
    for (int r = 0; r < 8; ++r) c[r] = -pcol;       // C[i][j] = -pred_j
    v2f b;
    b.x = (lane < 16) ? 1.0f : 0.0f;                // B row K=0 = ones; all else 0
    b.y = 0.0f;

    for (int t = 0; t < 128; ++t) {
        v2f a;
        a.x = lds_mz[t * 16 + n];   // A[M][K=0] = mz tile (lanes16-31 hit zero B rows)
        a.y = 0.0f;
        // D = A x B + C  ->  D[i][j] = mz_i - pred_j   (exact)
        v8f d = __builtin_amdgcn_wmma_f32_16x16x4_f32(
                    false, a, false, b, (short)0, c, false, false);
#pragma unroll
        for (int r = 0; r < 8; ++r) dmin[r] = fminf(dmin[r], fabsf(d[r]));
    }
#else
    const int hi = (lane >> 4) * 8;
    for (int t = 0; t < 128; ++t) {
#pragma unroll
        for (int r = 0; r < 8; ++r) {
            float m = lds_mz[t * 16 + hi + r];
            dmin[r] = fminf(dmin[r], fabsf(m - pcol));
        }
    }
#endif

    // min over the 8 accumulators (M = 0..7 / 8..15 per lane half) ...
    float red = dmin[0];
#pragma unroll
    for (int r = 1; r < 8; ++r) red = fminf(red, dmin[r]);
    // ... then across lane halves (combines the two M groups).
    red = fminf(red, __shfl_xor(red, 16));

    const int ig = g * 16 + n;
    float soft = (ig < plen) ? expf(-2.0f * red) : 0.0f;   // exp(-min/0.5); 0 if invalid
    float contrib = (lane < 16) ? soft * pred_inten[row * SROW + ig] : 0.0f;

    // deterministic wave sum
#pragma unroll
    for (int off = 16; off > 0; off >>= 1) contrib += __shfl_xor(contrib, off);
    if (lane == 0) s_wsum[wid] = contrib;
    __syncthreads();

    if (tid == 0) {
        float tot = 0.0f;
        for (int w = 0; w < 8; ++w) tot += s_wsum[w];      // fixed order
        ws[row * 16 + blockIdx.x] = tot;
    }
}

// ---------------------------------------------------------------------------
// Kernel 2: everything else (MSE terms, max-abs, sign penalty, row lengths,
// final loss assembly). One block, 1024 threads; 65536 elems per array.
// ---------------------------------------------------------------------------
__global__ void __launch_bounds__(1024)
finalize_kernel(const float* __restrict__ pred_mz,
                const float* __restrict__ mz,
                const float* __restrict__ inten,
                const float* __restrict__ pred_inten,
                const float* __restrict__ ws,
                float* __restrict__ out) {
    __shared__ int   s_plen[BROWS];
    __shared__ int   s_mlen[BROWS];
    __shared__ float s_part[32][4];
    __shared__ int   s_maxmz, s_maxit;   // float-as-int max (values >= 0)

    const int tid = threadIdx.x;
    if (tid < BROWS) { s_plen[tid] = 0; s_mlen[tid] = 0; }
    if (tid == 0)    { s_maxmz = 0; s_maxit = 0; }
    __syncthreads();

    float ssq_mz = 0.0f, ssq_it = 0.0f, sp_mz = 0.0f, sp_it = 0.0f;
    float mx_mz = 0.0f, mx_it = 0.0f;

    for (int idx = tid; idx < NTOT; idx += 1024) {
        float pm  = pred_mz[idx];
        float m   = mz[idx];
        float it  = inten[idx];
        float pit = pred_inten[idx];
        float d  = pm - m;   ssq_mz += d * d;
        float d2 = pit - it; ssq_it += d2 * d2;
        mx_mz = fmaxf(mx_mz, fabsf(m));
        mx_it = fmaxf(mx_it, fabsf(it));
        sp_mz += fabsf(sgnf(pm)  - sgnf(m));
        sp_it += fabsf(sgnf(pit) - sgnf(it));
        int row = idx >> 11, col = idx & (SROW - 1);
        if (pm < 0.0f) atomicMax(&s_plen[row], col);
        if (m  < 0.0f) atomicMax(&s_mlen[row], col);
    }

#pragma unroll
    for (int off = 16; off > 0; off >>= 1) {
        ssq_mz += __shfl_xor(ssq_mz, off);
        ssq_it += __shfl_xor(ssq_it, off);
        sp_mz  += __shfl_xor(sp_mz,  off);
        sp_it  += __shfl_xor(sp_it,  off);
        mx_mz = fmaxf(mx_mz, __shfl_xor(mx_mz, off));
        mx_it = fmaxf(mx_it, __shfl_xor(mx_it, off));
    }
    const int lane = tid & 31, wid = tid >> 5;
    if (lane == 0) {
        s_part[wid][0] = ssq_mz;
        s_part[wid][1] = ssq_it;
        s_part[wid][2] = sp_mz;
        s_part[wid][3] = sp_it;
        atomicMax(&s_maxmz, __float_as_int(mx_mz));
        atomicMax(&s_maxit, __float_as_int(mx_it));
    }
    __syncthreads();

    if (tid == 0) {
        float a = 0.0f, b = 0.0f, c = 0.0f, d = 0.0f;
        for (int w = 0; w < 32; ++w) {          // fixed order -> deterministic
            a += s_part[w][0]; b += s_part[w][1];
            c += s_part[w][2]; d += s_part[w][3];
        }
        float maxmz = __int_as_float(s_maxmz);
        float maxit = __int_as_float(s_maxit);

        float sj = 0.0f;
        for (int r = 0; r < BROWS; ++r) {
            float si = 0.0f;
            for (int k = 0; k < 16; ++k) si += ws[r * 16 + k];   // fixed order
            float uni = (float)(s_plen[r] + s_mlen[r]);
            sj += 1.0f - (si + 1e-8f) / (uni + 1e-8f);
        }
        const float invN = 1.0f / (float)NTOT;
        out[0] = sj / (float)BROWS;                     // SJ_W * sj_loss
        out[1] = (a * invN) / (maxmz * maxmz);          // MSE_MZ_W * loss_mz
        out[2] = (b * invN) / (maxit * maxit);          // MSE_INT_W * loss_intensity
        out[3] = 0.1f * 0.5f * (c * invN + d * invN);   // SIGN_W * sign_penalty
    }
}

extern "C" void kernel_launch(void* const* d_in, const int* in_sizes, int n_in,
                              void* d_out, int out_size, void* d_ws, size_t ws_size,
                              hipStream_t stream) {
    (void)in_sizes; (void)n_in; (void)out_size; (void)ws_size;
    const float* pred_mz        = (const float*)d_in[0];
    const float* mz             = (const float*)d_in[1];
    const float* intensity      = (const float*)d_in[2];
    const float* pred_intensity = (const float*)d_in[3];
    float* out = (float*)d_out;
    float* ws  = (float*)d_ws;    // 32*16 floats, fully rewritten every launch

    dim3 grid1(16, BROWS);
    sj_pairwise_kernel<<<grid1, 256, 0, stream>>>(pred_mz, mz, pred_intensity, ws);
    finalize_kernel<<<1, 1024, 0, stream>>>(pred_mz, mz, intensity, pred_intensity, ws, out);
}